// BagOfFeaturesModel_50620484550698
// MI455X (gfx1250) — compile-verified
//
#include <hip/hip_runtime.h>
#include <hip/hip_bf16.h>
#include <stdint.h>

// ---------------------------------------------------------------------------
// BagOfFeatures: gather -> threefry dropout (precomputed bitmask) ->
//                segment mean (counting sort) -> fp32 WMMA GEMM
// Target: gfx1250 (CDNA5, wave32, WMMA)
// ---------------------------------------------------------------------------

#define E_DIM 128
#define O_DIM 128

typedef float v2f __attribute__((ext_vector_type(2)));
typedef float v8f __attribute__((ext_vector_type(8)));

// ---------------- Threefry-2x32 (exact JAX semantics, key = (0, 42)) -------

__device__ __forceinline__ uint32_t rotl32(uint32_t x, int r) {
    return (x << r) | (x >> (32 - r));
}

__device__ __forceinline__ void threefry2x32_k42(uint32_t x0, uint32_t x1,
                                                 uint32_t& o0, uint32_t& o1) {
    const uint32_t k0 = 0u;
    const uint32_t k1 = 42u;
    const uint32_t k2 = k0 ^ k1 ^ 0x1BD11BDAu;
    x0 += k0; x1 += k1;
#define TF_R(r) { x0 += x1; x1 = rotl32(x1, (r)); x1 ^= x0; }
    TF_R(13) TF_R(15) TF_R(26) TF_R(6)   x0 += k1; x1 += k2 + 1u;
    TF_R(17) TF_R(29) TF_R(16) TF_R(24)  x0 += k2; x1 += k0 + 2u;
    TF_R(13) TF_R(15) TF_R(26) TF_R(6)   x0 += k0; x1 += k1 + 3u;
    TF_R(17) TF_R(29) TF_R(16) TF_R(24)  x0 += k1; x1 += k2 + 4u;
    TF_R(13) TF_R(15) TF_R(26) TF_R(6)   x0 += k2; x1 += k0 + 5u;
#undef TF_R
    o0 = x0; o1 = x1;
}

// ---------------- Phase A: dropout keep-bit mask ---------------------------
// JAX: bits over n = T*E elements; counter p pairs with p + n/2; out0 -> pos p,
// out1 -> pos p + n/2.  keep = uniform(bits) < 0.5  <=>  MSB(bits) == 0.
// One thread packs 32 consecutive pair indices -> one u32 for each half.
// (nhalf = T*64 is always a multiple of 32, so words align.)

__global__ void dropout_mask_kernel(uint32_t* __restrict__ mask, uint64_t nhalf) {
    const uint64_t gid = (uint64_t)blockIdx.x * blockDim.x + threadIdx.x;
    const uint64_t p0 = gid * 32ull;
    if (p0 >= nhalf) return;
    uint32_t w0 = 0u, w1 = 0u;
#pragma unroll 4
    for (int i = 0; i < 32; ++i) {
        uint32_t o0, o1;
        threefry2x32_k42((uint32_t)(p0 + (uint64_t)i),
                         (uint32_t)(p0 + (uint64_t)i + nhalf), o0, o1);
        w0 |= ((~o0) >> 31) << i;   // keep iff MSB == 0
        w1 |= ((~o1) >> 31) << i;
    }
    mask[gid] = w0;
    mask[gid + (nhalf >> 5)] = w1;
}

// ---------------- Phase 0: zero segment counts -----------------------------

__global__ void zero_counts_kernel(uint32_t* __restrict__ counts, int S) {
    int i = blockIdx.x * blockDim.x + threadIdx.x;
    if (i < S) counts[i] = 0u;
}

// ---------------- Phase 1: histogram of segment ids ------------------------

__global__ void count_kernel(const int* __restrict__ indices,
                             uint32_t* __restrict__ counts, int T) {
    int t = blockIdx.x * blockDim.x + threadIdx.x;
    if (t < T) atomicAdd(&counts[indices[t]], 1u);
}

// ---------------- Phase 2: exclusive scan (S <= 16384, one workgroup) ------

__global__ void scan_kernel(const uint32_t* __restrict__ counts,
                            uint32_t* __restrict__ offsets,
                            uint32_t* __restrict__ cursor, int S) {
    __shared__ uint32_t part[1024];
    const int tid = threadIdx.x;
    const int base = tid * 16;
    uint32_t local[16];
    uint32_t sum = 0u;
#pragma unroll
    for (int i = 0; i < 16; ++i) {
        uint32_t v = (base + i < S) ? counts[base + i] : 0u;
        local[i] = v;
        sum += v;
    }
    part[tid] = sum;
    __syncthreads();
    for (int off = 1; off < 1024; off <<= 1) {
        uint32_t v = (tid >= off) ? part[tid - off] : 0u;
        __syncthreads();
        part[tid] += v;
        __syncthreads();
    }
    uint32_t run = (tid > 0) ? part[tid - 1] : 0u;  // exclusive prefix
#pragma unroll
    for (int i = 0; i < 16; ++i) {
        if (base + i < S) {
            offsets[base + i] = run;
            cursor[base + i]  = run;
        }
        run += local[i];
    }
}

// ---------------- Phase 3: scatter ids into segment-sorted order -----------

__global__ void scatter_kernel(const int* __restrict__ indices,
                               uint32_t* __restrict__ cursor,
                               uint32_t* __restrict__ sorted_t, int T) {
    int t = blockIdx.x * blockDim.x + threadIdx.x;
    if (t < T) {
        uint32_t pos = atomicAdd(&cursor[indices[t]], 1u);
        sorted_t[pos] = (uint32_t)t;
    }
}

// ---------------- Phase 4: per-segment gather + masked mean ----------------
// One wave per segment; lane owns 4 consecutive columns (float4 gather =
// coalesced 512B/wave from the L2-resident table).  Dropout mask: one u32
// word per 8 lanes (16B/wave), 4-bit nibble per lane.  The 1/(1-p)=2 scale
// is folded into the final mean divide.

__global__ void segment_mean_kernel(const int* __restrict__ features,
                                    const uint32_t* __restrict__ sorted_t,
                                    const uint32_t* __restrict__ offsets,
                                    const uint32_t* __restrict__ counts,
                                    const uint32_t* __restrict__ mask,
                                    const float* __restrict__ emb,
                                    float* __restrict__ mean, int S) {
    const int wavesPerBlock = blockDim.x >> 5;
    const int seg = blockIdx.x * wavesPerBlock + ((int)threadIdx.x >> 5);
    if (seg >= S) return;
    const int lane = threadIdx.x & 31;
    const int e0 = lane * 4;
    const int mword_off = lane >> 3;          // which u32 of the 4-word row mask
    const int mshift = (lane & 7) * 4;        // nibble position inside the word

    const uint32_t start = offsets[seg];
    const uint32_t cnt   = counts[seg];

    float4 acc = make_float4(0.f, 0.f, 0.f, 0.f);
    for (uint32_t i = 0; i < cnt; ++i) {
        const uint32_t t = sorted_t[start + i];
        const int f = features[t];
        const float4 v = *(const float4*)(emb + (size_t)f * E_DIM + e0);
        const uint32_t nib = mask[(size_t)t * (E_DIM / 32) + mword_off] >> mshift;
        acc.x += (nib & 1u) ? v.x : 0.0f;
        acc.y += (nib & 2u) ? v.y : 0.0f;
        acc.z += (nib & 4u) ? v.z : 0.0f;
        acc.w += (nib & 8u) ? v.w : 0.0f;
    }
    const float inv = 2.0f / fmaxf((float)cnt, 1.0f);   // dropout scale folded in
    float4 m = make_float4(acc.x * inv, acc.y * inv, acc.z * inv, acc.w * inv);
    *(float4*)(mean + (size_t)seg * E_DIM + e0) = m;
}

// ---------------- Phase 5: out = mean @ W^T + b via V_WMMA_F32_16X16X4_F32 -
// One wave per 16x16 output tile; K=128 consumed in 32 steps of 4.
// A (16x4 f32): lanes 0-15 carry K=k,k+1 for M=lane; lanes 16-31 carry K=k+2,k+3.
// B (4x16 f32): v0 = rows k / k+2, v1 = k+1 / k+3.
// C/D (16x16 f32, 8 VGPRs): vr -> M = r + (lane>=16)*8, N = lane&15.

__global__ void gemm_wmma_kernel(const float* __restrict__ mean,
                                 const float* __restrict__ W,
                                 const float* __restrict__ bias,
                                 float* __restrict__ out, int S) {
    const int wavesPerBlock = blockDim.x >> 5;
    const int waveId = blockIdx.x * wavesPerBlock + ((int)threadIdx.x >> 5);
    const int ntile_n = O_DIM / 16;                 // 8
    const int tile_m = waveId / ntile_n;
    const int tile_n = waveId % ntile_n;
    if (tile_m * 16 >= S) return;                   // wave-uniform: EXEC stays full

    const int lane = threadIdx.x & 31;
    const int half = lane >> 4;                     // 0: K+0/K+1, 1: K+2/K+3
    const int l15  = lane & 15;

    const float* Arow = mean + (size_t)(tile_m * 16 + l15) * E_DIM;  // mean[m][k]
    const float* Brow = W    + (size_t)(tile_n * 16 + l15) * E_DIM;  // W[n][k] = B[k][n]

    v8f c = {};
#pragma unroll 4
    for (int k = 0; k < E_DIM; k += 4) {
        const v2f a = *(const v2f*)(Arow + k + half * 2);
        const v2f b = *(const v2f*)(Brow + k + half * 2);
        c = __builtin_amdgcn_wmma_f32_16x16x4_f32(
                /*neg_a=*/false, a, /*neg_b=*/false, b,
                /*c_mod=*/(short)0, c, /*reuse_a=*/false, /*reuse_b=*/false);
    }

    const float bv = bias[tile_n * 16 + l15];
#pragma unroll
    for (int r = 0; r < 8; ++r) {
        out[(size_t)(tile_m * 16 + r + half * 8) * O_DIM + tile_n * 16 + l15] =
            c[r] + bv;
    }
}

// ---------------------------------------------------------------------------

static inline size_t align_up(size_t x, size_t a) { return (x + a - 1) & ~(a - 1); }

extern "C" void kernel_launch(void* const* d_in, const int* in_sizes, int n_in,
                              void* d_out, int out_size, void* d_ws, size_t ws_size,
                              hipStream_t stream) {
    const int*   features = (const int*)d_in[0];
    const int*   indices  = (const int*)d_in[1];
    // d_in[2] = num_segments (device scalar; S derived from out_size instead)
    const float* emb      = (const float*)d_in[3];
    const float* W        = (const float*)d_in[4];
    const float* bias     = (const float*)d_in[5];
    float*       out      = (float*)d_out;

    const int T = in_sizes[0];
    const int S = out_size / O_DIM;
    const uint64_t nhalf = ((uint64_t)T * (uint64_t)E_DIM) / 2ull;  // multiple of 32

    // workspace carve-out
    char* ws = (char*)d_ws;
    size_t off = 0;
    uint32_t* counts   = (uint32_t*)(ws + off); off += align_up((size_t)S * 4, 256);
    uint32_t* offsets  = (uint32_t*)(ws + off); off += align_up((size_t)S * 4, 256);
    uint32_t* cursor   = (uint32_t*)(ws + off); off += align_up((size_t)S * 4, 256);
    uint32_t* sorted_t = (uint32_t*)(ws + off); off += align_up((size_t)T * 4, 256);
    uint32_t* maskbuf  = (uint32_t*)(ws + off); off += align_up((size_t)T * (E_DIM / 8), 256);
    float*    mean     = (float*)(ws + off);
    (void)ws_size; (void)n_in;

    const int TB = 256;

    // dropout mask: nhalf/32 packer threads
    {
        const uint64_t nthreads = nhalf >> 5;
        const uint32_t blocks = (uint32_t)((nthreads + TB - 1) / TB);
        dropout_mask_kernel<<<blocks, TB, 0, stream>>>(maskbuf, nhalf);
    }

    zero_counts_kernel<<<(S + TB - 1) / TB, TB, 0, stream>>>(counts, S);
    count_kernel<<<(T + TB - 1) / TB, TB, 0, stream>>>(indices, counts, T);
    scan_kernel<<<1, 1024, 0, stream>>>(counts, offsets, cursor, S);
    scatter_kernel<<<(T + TB - 1) / TB, TB, 0, stream>>>(indices, cursor, sorted_t, T);

    // one wave per segment, 8 waves per block
    segment_mean_kernel<<<(S + 7) / 8, TB, 0, stream>>>(
        features, sorted_t, offsets, counts, maskbuf, emb, mean, S);

    // one wave per 16x16 tile; (S/16)*(O/16) tiles, 8 waves per block
    const int tiles = ((S + 15) / 16) * (O_DIM / 16);
    gemm_wmma_kernel<<<(tiles + 7) / 8, TB, 0, stream>>>(mean, W, bias, out, S);
}